// LlamaDecodeAttention_79439715106872
// MI455X (gfx1250) — compile-verified
//
#include <hip/hip_runtime.h>
#include <hip/hip_bf16.h>

typedef float v2f __attribute__((ext_vector_type(2)));
typedef float v8f __attribute__((ext_vector_type(8)));

#define B_    64
#define KV_   2048
#define HID_  4096
#define H_    32
#define KVH_  8
#define D_    128
#define G_    4
#define QKV_N ((H_ + 2 * KVH_) * D_)   /* 6144 */

// -------------------------------------------------------------------------
// C[M x N] = A[M x K] (row major) @ W[K x N] (row major) + bias[N]
// Split-K x4: each 16x16 tile is computed by 4 waves (one wave32 per K/4
// chunk) via V_WMMA_F32_16X16X4_F32; partials reduced through LDS in a
// fixed order (deterministic). One block = 8 waves = 2 tiles x 4 splits.
// M,N,K are template constants so the K-loop has a literal trip count and
// all tile index math folds to shifts. The wave id is readfirstlane'd so
// tile/split/kBeg are SGPRs -> scalar loop control, EXEC all-1s for WMMA,
// and the unroll-16 load clause (48 loads, ~8KB/wave in flight) survives.
// Fragment layouts per CDNA5 ISA 7.12.2:
//   A 16x4 : lane(0-15)=M, half selects K pair: v0=K(2h), v1=K(2h+1)
//   B 4x16 : lane%16=N,   v0=row K(2h), v1=row K(2h+1)
//   C 16x16: vgpr i -> row m0+i+8*half, col n0+lane%16
// -------------------------------------------------------------------------
template <int M, int N, int K>
__global__ __launch_bounds__(256) void gemm_bias_wmma_splitk(
    const float* __restrict__ A, const float* __restrict__ W,
    const float* __restrict__ bias, float* __restrict__ C)
{
    const int tid  = threadIdx.x;
    const int wave = __builtin_amdgcn_readfirstlane(tid >> 5);  // uniform SGPR
    const int lane = tid & 31;
    const int half = lane >> 4;          // 0 or 1
    const int lm   = lane & 15;

    constexpr int mTiles = M / 16;
    constexpr int nTiles = N / 16;
    constexpr int totalTiles = mTiles * nTiles;
    constexpr int kLen = K / 4;          // K-range per split (1024)

    const int tile  = blockIdx.x * 2 + (wave >> 2);   // 2 tiles per block
    const int split = wave & 3;                       // 4 K-splits per tile

    __shared__ float s_part[8][256];                  // 8 KB partials

    v8f acc = {};
    if (tile < totalTiles) {                          // scalar branch
        const int mt = tile % mTiles;
        const int nt = tile / mTiles;
        const int m0 = mt << 4;
        const int n0 = nt << 4;
        const int kBeg = split * kLen;

        const float* Arow = A + (size_t)(m0 + lm) * K + kBeg + 2 * half;
        const float* Wcol = W + (size_t)(kBeg + 2 * half) * N + n0 + lm;

        #pragma unroll 16
        for (int kk = 0; kk < kLen; kk += 4) {
            v2f a = *(const v2f*)(Arow + kk);         // A[m][kB+kk+2h], +1
            v2f b;
            b.x = Wcol[(size_t)kk * N];               // W[kB+kk+2h  ][n]
            b.y = Wcol[(size_t)(kk + 1) * N];         // W[kB+kk+2h+1][n]
            acc = __builtin_amdgcn_wmma_f32_16x16x4_f32(
                false, a, false, b, (short)0, acc, false, false);
        }
    }

    #pragma unroll
    for (int i = 0; i < 8; ++i) s_part[wave][i * 32 + lane] = acc[i];
    __syncthreads();

    // Reduce the 4 K-splits per tile and write out with bias.
    const int tsel  = tid >> 7;                       // which of the 2 tiles
    const int rtile = blockIdx.x * 2 + tsel;
    if (rtile >= totalTiles) return;
    const int mt = rtile % mTiles;
    const int nt = rtile / mTiles;
    const int m0 = mt << 4;
    const int n0 = nt << 4;
    const int jb = tid & 127;

    #pragma unroll
    for (int rep = 0; rep < 2; ++rep) {
        const int j = jb + rep * 128;                 // element id = i*32 + lane
        const float s = s_part[tsel * 4 + 0][j] + s_part[tsel * 4 + 1][j]
                      + s_part[tsel * 4 + 2][j] + s_part[tsel * 4 + 3][j];
        const int i   = j >> 5;
        const int ln  = j & 31;
        const int hh  = ln >> 4;
        const int lmm = ln & 15;
        const int row = m0 + i + 8 * hh;
        const int col = n0 + lmm;
        C[(size_t)row * N + col] = s + bias[col];
    }
}

// -------------------------------------------------------------------------
// Neox-style RoPE in-place on q (heads 0..31) and k_new (heads 32..39)
// inside the qkv workspace. blockDim = 64 (one thread per rotation pair).
// -------------------------------------------------------------------------
__global__ __launch_bounds__(64) void rope_kernel(
    float* __restrict__ qkv, const int* __restrict__ positions)
{
    const int hh = blockIdx.x % (H_ + KVH_);
    const int b  = blockIdx.x / (H_ + KVH_);
    const int j  = threadIdx.x;                       // 0..63 (half of D)

    const float pos      = (float)positions[b];
    // inv_freq = theta^(-2j/D) = exp(-j * 2*ln(theta)/D)
    const float inv_freq = __expf(-(float)j * (9.210340371976184f * 2.0f / (float)D_));
    const float ang = pos * inv_freq;
    const float c = cosf(ang);
    const float s = sinf(ang);

    const size_t base = (size_t)b * QKV_N +
        (hh < H_ ? (size_t)hh * D_ : (size_t)H_ * D_ + (size_t)(hh - H_) * D_);

    const float x1 = qkv[base + j];
    const float x2 = qkv[base + j + 64];
    qkv[base + j]      = x1 * c - x2 * s;
    qkv[base + j + 64] = x2 * c + x1 * s;
}

// -------------------------------------------------------------------------
// GQA decode attention: one workgroup (256 threads, 8 waves) per (b, kvh).
// Streams K rows with per-lane contiguous float4 (full cachelines), scores
// in LDS, wave32 shuffle softmax, then coalesced float2 V streaming for PV.
// -------------------------------------------------------------------------
__global__ __launch_bounds__(256) void attn_kernel(
    const float* __restrict__ qkv, const float* __restrict__ kc,
    const float* __restrict__ vc, float* __restrict__ out)
{
    const int b   = blockIdx.x / KVH_;
    const int kvh = blockIdx.x % KVH_;
    const int t   = threadIdx.x;

    __shared__ __align__(16) float s_q[G_][D_];       // 2 KB (pre-scaled)
    __shared__ float s_sc[G_][KV_ + 16];              // scores/probs, ~33 KB
    __shared__ float s_inv[G_];                       // 1/sum per group

    const float scale = 0.08838834764831845f;         // D^-0.5

    for (int i = t; i < G_ * D_; i += 256) {
        const int g = i / D_, d = i % D_;
        s_q[g][d] = qkv[(size_t)b * QKV_N + (size_t)(kvh * G_ + g) * D_ + d] * scale;
    }
    __syncthreads();

    // ---- phase 1: scores[g][s] = (q_g . k_s) * scale, s in [0, KV] ----
    const float* kbase = kc + ((size_t)b * KV_ * KVH_ + kvh) * D_;
    const float* knew  = qkv + (size_t)b * QKV_N + (size_t)H_ * D_ + (size_t)kvh * D_;
    for (int s = t; s <= KV_; s += 256) {
        const float* krow = (s < KV_) ? (kbase + (size_t)s * (KVH_ * D_)) : knew;
        float a0 = 0.f, a1 = 0.f, a2 = 0.f, a3 = 0.f;
        #pragma unroll 8
        for (int d = 0; d < D_; d += 4) {
            const float4 kk = *(const float4*)(krow + d);
            const float4 q0 = *(const float4*)(&s_q[0][d]);
            const float4 q1 = *(const float4*)(&s_q[1][d]);
            const float4 q2 = *(const float4*)(&s_q[2][d]);
            const float4 q3 = *(const float4*)(&s_q[3][d]);
            a0 += kk.x * q0.x + kk.y * q0.y + kk.z * q0.z + kk.w * q0.w;
            a1 += kk.x * q1.x + kk.y * q1.y + kk.z * q1.z + kk.w * q1.w;
            a2 += kk.x * q2.x + kk.y * q2.y + kk.z * q2.z + kk.w * q2.w;
            a3 += kk.x * q3.x + kk.y * q3.y + kk.z * q3.z + kk.w * q3.w;
        }
        s_sc[0][s] = a0; s_sc[1][s] = a1; s_sc[2][s] = a2; s_sc[3][s] = a3;
    }
    __syncthreads();

    // ---- phase 2: softmax stats; wave g handles group g (wave32 shuffles) ----
    const int wave = t >> 5, lane = t & 31;
    if (wave < G_) {
        const int g = wave;
        float m = -3.4e38f;
        for (int s = lane; s <= KV_; s += 32) m = fmaxf(m, s_sc[g][s]);
        #pragma unroll
        for (int o = 16; o > 0; o >>= 1) m = fmaxf(m, __shfl_xor(m, o, 32));
        float sum = 0.f;
        for (int s = lane; s <= KV_; s += 32) {
            const float e = expf(s_sc[g][s] - m);
            s_sc[g][s] = e;
            sum += e;
        }
        #pragma unroll
        for (int o = 16; o > 0; o >>= 1) sum += __shfl_xor(sum, o, 32);
        if (lane == 0) s_inv[g] = 1.0f / sum;
    }
    __syncthreads();

    // ---- phase 3: attn[g][d] = sum_s p[g][s] * v[s][d] (coalesced over d) ----
    const int g = t >> 6;            // 0..3
    const int d = (t & 63) * 2;      // 0,2,...,126
    const float* vbase = vc + ((size_t)b * KV_ * KVH_ + kvh) * D_ + d;
    float ax = 0.f, ay = 0.f;
    #pragma unroll 16
    for (int s = 0; s < KV_; ++s) {
        const float2 vv = *(const float2*)(vbase + (size_t)s * (KVH_ * D_));
        const float p = s_sc[g][s];
        ax += p * vv.x;
        ay += p * vv.y;
    }
    {   // new token's value
        const float* vn = qkv + (size_t)b * QKV_N + (size_t)(H_ + KVH_) * D_ +
                          (size_t)kvh * D_ + d;
        const float p = s_sc[g][KV_];
        ax += p * vn[0];
        ay += p * vn[1];
    }
    const float inv = s_inv[g];
    const size_t ob = (size_t)b * HID_ + (size_t)(kvh * G_ + g) * D_ + d;
    out[ob]     = ax * inv;
    out[ob + 1] = ay * inv;
}

// -------------------------------------------------------------------------
extern "C" void kernel_launch(void* const* d_in, const int* in_sizes, int n_in,
                              void* d_out, int out_size, void* d_ws, size_t ws_size,
                              hipStream_t stream)
{
    const float* hs   = (const float*)d_in[0];   // [64, 4096]
    const float* kc   = (const float*)d_in[1];   // [64, 2048, 8, 128]
    const float* vc   = (const float*)d_in[2];   // [64, 2048, 8, 128]
    const int*   pos  = (const int*)  d_in[3];   // [64]
    const float* Wqkv = (const float*)d_in[4];   // [4096, 6144]
    const float* bqkv = (const float*)d_in[5];   // [6144]
    const float* Wo   = (const float*)d_in[6];   // [4096, 4096]
    const float* bo   = (const float*)d_in[7];   // [4096]
    float* out  = (float*)d_out;                 // [64, 4096]

    float* qkv  = (float*)d_ws;                  // [64, 6144]
    float* attn = qkv + (size_t)B_ * QKV_N;      // [64, 4096]

    // 1) QKV projection: tiles = 4*384 = 1536; 2 tiles x 4 splits per block -> 768 blocks
    gemm_bias_wmma_splitk<B_, QKV_N, HID_><<<768, 256, 0, stream>>>(hs, Wqkv, bqkv, qkv);

    // 2) RoPE on q and k_new (in-place in qkv workspace)
    rope_kernel<<<B_ * (H_ + KVH_), 64, 0, stream>>>(qkv, pos);

    // 3) Attention: one WG per (b, kv head)
    attn_kernel<<<B_ * KVH_, 256, 0, stream>>>(qkv, kc, vc, attn);

    // 4) Output projection: tiles = 4*256 = 1024 -> 512 blocks
    gemm_bias_wmma_splitk<B_, HID_, HID_><<<512, 256, 0, stream>>>(attn, Wo, bo, out);
}